// MultiHopGNN_45664092291084
// MI455X (gfx1250) — compile-verified
//
#include <hip/hip_runtime.h>
#include <hip/hip_bf16.h>

// ---------------- Problem constants (match reference) ----------------
#define N_NODES   16384
#define N_EDGES   262144
#define IN_DIM    64
#define HIDDEN    256
#define NCLS      40
#define NCLS_PAD  48                       // 3 WMMA N-tiles
#define BM_WPR    512                      // bitmap words per row = 16384/32
#define BM_WORDS  (N_NODES * BM_WPR)       // 8,388,608 u32 = 32 MB

// LDS pair-row strides (float2 units). Need 2*stride % 64 == 32 so the two
// lane-halves (pair-rows p and p+1) hit disjoint 32-bank halves.
#define G1_LDS_STRIDE 80                   // 96 pair-rows * 80 * 8B = 61,440 B
#define G2_LDS_STRIDE 48                   // 128 pair-rows * 48 * 8B = 49,152 B

typedef float v2f __attribute__((ext_vector_type(2)));
typedef float v8f __attribute__((ext_vector_type(8)));

// ---------------- Stage 0: zero the adjacency bitmap ----------------
__global__ void k_zero_bitmap(unsigned* __restrict__ bm) {
    int idx = blockIdx.x * blockDim.x + threadIdx.x;
    if (idx < BM_WORDS) bm[idx] = 0u;
}

// ---------------- Stage 1: scatter edges into bitmap (dedup) ----------------
__global__ void k_set_bits(const int* __restrict__ ei, unsigned* __restrict__ bm) {
    int e = blockIdx.x * blockDim.x + threadIdx.x;
    if (e >= N_EDGES) return;
    int i = ei[e];             // row   (edge_index[0])
    int j = ei[N_EDGES + e];   // col   (edge_index[1])
    size_t bitpos = (size_t)i * N_NODES + (size_t)j;
    atomicOr(&bm[bitpos >> 5], 1u << (bitpos & 31u));
}

// ---------------- Stage 2: degree via popcount, dinv = (deg+eps)^-1/2 ----------------
__global__ void k_dinv(const unsigned* __restrict__ bm, float* __restrict__ dinv) {
    int wavesPerBlk = blockDim.x >> 5;
    int row  = blockIdx.x * wavesPerBlk + (threadIdx.x >> 5);
    int lane = threadIdx.x & 31;
    if (row >= N_NODES) return;
    const unsigned* brow = bm + (size_t)row * BM_WPR;
    int cnt = 0;
    #pragma unroll
    for (int w = 0; w < BM_WPR; w += 32) cnt += __popc(brow[w + lane]);
    #pragma unroll
    for (int off = 16; off > 0; off >>= 1) cnt += __shfl_down(cnt, off, 32);
    if (lane == 0) dinv[row] = 1.0f / sqrtf((float)cnt + 1e-8f);
}

// ---------------- Stage 3: Fs[j,:] = dinv[j] * F[j,:] ----------------
__global__ void k_prescale(const float* __restrict__ F, const float* __restrict__ dinv,
                           float* __restrict__ Fs) {
    int idx = blockIdx.x * blockDim.x + threadIdx.x;
    if (idx >= N_NODES * IN_DIM) return;
    Fs[idx] = dinv[idx >> 6] * F[idx];
}

// ---------------- Stage 4: deterministic bitmap-gather hop ----------------
// Y[i,:] = dinv[i] * sum_{j in row i} Fs[j,:]   (one wave per row, lane owns 2 channels)
__global__ void k_gather(const unsigned* __restrict__ bm, const float* __restrict__ Fs,
                         const float* __restrict__ dinv, float* __restrict__ Y) {
    int wavesPerBlk = blockDim.x >> 5;
    int row  = blockIdx.x * wavesPerBlk + (threadIdx.x >> 5);
    int lane = threadIdx.x & 31;
    if (row >= N_NODES) return;
    const unsigned* brow = bm + (size_t)row * BM_WPR;
    float a0 = 0.0f, a1 = 0.0f;
    for (int w = 0; w < BM_WPR; ++w) {
        unsigned bits = brow[w];
        while (bits) {
            int j = (w << 5) + __builtin_ctz(bits);
            bits &= bits - 1u;
            const float* fr = Fs + (size_t)j * IN_DIM;
            a0 += fr[lane];
            a1 += fr[lane + 32];
        }
    }
    float s = dinv[row];
    Y[(size_t)row * IN_DIM + lane]      = s * a0;
    Y[(size_t)row * IN_DIM + lane + 32] = s * a1;
}

// ---------------- Stage 5: GEMM1 + bias + ReLU via V_WMMA_F32_16X16X4_F32 ----------------
// H = [X | AX | A2X] (16384 x 192), Hmid = relu(H @ W1 + b1)  (16384 x 256)
// Block = 8 waves sharing one 64-col W1 slab in LDS (pair-row float2 layout,
// bank-conflict-free ds_load_b64 B-fragments). Wave -> one M-tile, 4 C-tiles.
__global__ void k_gemm1(const float* __restrict__ X,  const float* __restrict__ AX,
                        const float* __restrict__ A2X, const float* __restrict__ W1,
                        const float* __restrict__ b1,  float* __restrict__ Hmid) {
    __shared__ float2 sW[96 * G1_LDS_STRIDE];            // 192 K-rows as 96 pairs
    int tid    = threadIdx.x;
    int waveId = tid >> 5;
    int lane   = tid & 31;
    int colb   = blockIdx.y * 64;                        // 4 column groups
    int mtile  = blockIdx.x * 8 + waveId;                // 1024 M-tiles
    int nl     = lane & 15;
    int half   = lane >> 4;
    int arow   = mtile * 16 + nl;                        // A-frag row for this lane

    // Cooperative stage: sW[p][c] = ( W1[2p][colb+c], W1[2p+1][colb+c] )
    for (int idx = tid; idx < 96 * 64; idx += 256) {
        int p = idx >> 6, c = idx & 63;
        const float* w = W1 + (size_t)(2 * p) * HIDDEN + colb + c;
        sW[p * G1_LDS_STRIDE + c] = make_float2(w[0], w[HIDDEN]);
    }
    __syncthreads();

    v8f acc[4];
    #pragma unroll
    for (int t = 0; t < 4; ++t) {
        float bv = b1[colb + 16 * t + nl];
        acc[t] = (v8f){bv, bv, bv, bv, bv, bv, bv, bv};
    }

    #pragma unroll
    for (int seg = 0; seg < 3; ++seg) {
        const float* F = (seg == 0) ? X : (seg == 1) ? AX : A2X;
        const float* Fr = F + (size_t)arow * IN_DIM;
        #pragma unroll
        for (int k = 0; k < IN_DIM; k += 4) {
            // A frag: lanes 0-15 hold K = k,k+1 ; lanes 16-31 hold K = k+2,k+3
            v2f a = *(const v2f*)(Fr + k + 2 * half);
            // B frag: pair-row p = seg*32 + k/2 + half -> (W[krow][c], W[krow+1][c])
            const float2* bp = sW + (seg * 32 + (k >> 1) + half) * G1_LDS_STRIDE + nl;
            #pragma unroll
            for (int t = 0; t < 4; ++t) {
                float2 bb = bp[16 * t];                  // one ds_load_b64
                v2f b; b.x = bb.x; b.y = bb.y;
                acc[t] = __builtin_amdgcn_wmma_f32_16x16x4_f32(
                    false, a, false, b, (short)0, acc[t], false, false);
            }
        }
    }

    #pragma unroll
    for (int t = 0; t < 4; ++t) {
        int col = colb + 16 * t + nl;
        #pragma unroll
        for (int r = 0; r < 8; ++r) {
            int row = mtile * 16 + half * 8 + r;
            float v = acc[t][r];
            Hmid[(size_t)row * HIDDEN + col] = v > 0.0f ? v : 0.0f;
        }
    }
}

// ---------------- Stage 6: GEMM2 + bias (N=40, zero-padded to 48 in LDS) ----------------
__global__ void k_gemm2(const float* __restrict__ Hmid, const float* __restrict__ W2,
                        const float* __restrict__ b2, float* __restrict__ Out) {
    __shared__ float2 sW[128 * G2_LDS_STRIDE];           // 256 K-rows as 128 pairs, 48 cols
    int tid    = threadIdx.x;
    int waveId = tid >> 5;
    int lane   = tid & 31;
    int mtile  = blockIdx.x * 8 + waveId;                // 1024 M-tiles
    int nl     = lane & 15;
    int half   = lane >> 4;
    int arow   = mtile * 16 + nl;
    const float* Hr = Hmid + (size_t)arow * HIDDEN;

    // Cooperative stage with zero padding for cols 40..47 (inner loop guard-free)
    for (int idx = tid; idx < 128 * NCLS_PAD; idx += 256) {
        int p = idx / NCLS_PAD, c = idx % NCLS_PAD;
        float2 v = make_float2(0.0f, 0.0f);
        if (c < NCLS) {
            const float* w = W2 + (size_t)(2 * p) * NCLS + c;
            v = make_float2(w[0], w[NCLS]);
        }
        sW[p * G2_LDS_STRIDE + c] = v;
    }
    __syncthreads();

    v8f acc[3];
    #pragma unroll
    for (int t = 0; t < 3; ++t) {
        int col = 16 * t + nl;
        float bv = (col < NCLS) ? b2[col] : 0.0f;
        acc[t] = (v8f){bv, bv, bv, bv, bv, bv, bv, bv};
    }

    #pragma unroll 8
    for (int k = 0; k < HIDDEN; k += 4) {
        v2f a = *(const v2f*)(Hr + k + 2 * half);
        const float2* bp = sW + ((k >> 1) + half) * G2_LDS_STRIDE + nl;
        #pragma unroll
        for (int t = 0; t < 3; ++t) {
            float2 bb = bp[16 * t];
            v2f b; b.x = bb.x; b.y = bb.y;
            acc[t] = __builtin_amdgcn_wmma_f32_16x16x4_f32(
                false, a, false, b, (short)0, acc[t], false, false);
        }
    }

    #pragma unroll
    for (int t = 0; t < 3; ++t) {
        int col = 16 * t + nl;
        if (col < NCLS) {
            #pragma unroll
            for (int r = 0; r < 8; ++r) {
                int row = mtile * 16 + half * 8 + r;
                Out[(size_t)row * NCLS + col] = acc[t][r];
            }
        }
    }
}

// ---------------- Host-side orchestration ----------------
extern "C" void kernel_launch(void* const* d_in, const int* in_sizes, int n_in,
                              void* d_out, int out_size, void* d_ws, size_t ws_size,
                              hipStream_t stream) {
    const float* X  = (const float*)d_in[0];
    const int*   EI = (const int*)  d_in[1];
    const float* W1 = (const float*)d_in[2];
    const float* b1 = (const float*)d_in[3];
    const float* W2 = (const float*)d_in[4];
    const float* b2 = (const float*)d_in[5];
    float* Out = (float*)d_out;

    // workspace layout
    unsigned* bm   = (unsigned*)d_ws;                          // 32 MB bitmap
    float*    dinv = (float*)((char*)d_ws + (size_t)BM_WORDS * 4);
    float*    Fs   = dinv + N_NODES;                           // 4 MB
    float*    AX   = Fs   + (size_t)N_NODES * IN_DIM;          // 4 MB
    float*    A2X  = AX   + (size_t)N_NODES * IN_DIM;          // 4 MB
    float*    Hmid = A2X  + (size_t)N_NODES * IN_DIM;          // 16 MB

    // 0) rebuild bitmap every call (ws poisoned once; we rewrite all we read)
    k_zero_bitmap<<<BM_WORDS / 256, 256, 0, stream>>>(bm);
    // 1) dedup scatter (integer atomicOr -> order-independent, deterministic)
    k_set_bits<<<N_EDGES / 256, 256, 0, stream>>>(EI, bm);
    // 2) degrees -> dinv
    k_dinv<<<N_NODES / 8, 256, 0, stream>>>(bm, dinv);
    // 3) hop 1: AX = Anorm @ X
    k_prescale<<<(N_NODES * IN_DIM) / 256, 256, 0, stream>>>(X, dinv, Fs);
    k_gather<<<N_NODES / 8, 256, 0, stream>>>(bm, Fs, dinv, AX);
    // 4) hop 2: A2X = Anorm @ AX
    k_prescale<<<(N_NODES * IN_DIM) / 256, 256, 0, stream>>>(AX, dinv, Fs);
    k_gather<<<N_NODES / 8, 256, 0, stream>>>(bm, Fs, dinv, A2X);
    // 5) Hmid = relu([X|AX|A2X] @ W1 + b1)   (WMMA f32, W1 slab in LDS)
    k_gemm1<<<dim3(N_NODES / 128, 4), 256, 0, stream>>>(X, AX, A2X, W1, b1, Hmid);
    // 6) Out = Hmid @ W2 + b2                 (WMMA f32, W2 in LDS)
    k_gemm2<<<N_NODES / 128, 256, 0, stream>>>(Hmid, W2, b2, Out);
}